// SSM_Eagle_87986700026023
// MI455X (gfx1250) — compile-verified
//
#include <hip/hip_runtime.h>
#include <stdint.h>

// Problem constants (from reference setup)
#define B_      8
#define N_      64
#define V_      128000
#define K_      64
#define BK_     (B_ * K_)

// Select machinery
#define NBINS       8192      // bin = float_bits >> 18 (exp + 5 mantissa bits)
#define BIN_SHIFT   18
#define CAP         4096      // candidate capacity per batch row

// Streaming tiles for async global->LDS
#define TILE_FLOATS 1024      // 256 threads x float4
#define TILE_BYTES  4096
#define NTILES      (V_ / TILE_FLOATS)   // 125 exactly

// Workspace layout (bytes)
#define HIST_BYTES  (B_ * NBINS * 4)     // 256 KB
#define CNT_OFF     HIST_BYTES
#define THR_OFF     (HIST_BYTES + 64)
#define CAND_OFF    (HIST_BYTES + 128)   // uint2[B_*CAP] -> 256 KB

__device__ __forceinline__ void async_load_b128(unsigned lds_addr, unsigned goff,
                                                const float* gbase) {
    // GVS mode: mem = SGPR64 + VGPR32 offset; vdst = per-lane LDS byte address
    asm volatile("global_load_async_to_lds_b128 %0, %1, %2"
                 :: "v"(lds_addr), "v"(goff), "s"(gbase)
                 : "memory");
}

// ---------------------------------------------------------------------------
// Pass 1: per-(b,n) block streams V floats through LDS (async double buffer),
// builds LDS histogram of (prob bits >> 18), merges into global histogram.
// ---------------------------------------------------------------------------
__global__ __launch_bounds__(256) void hist_kernel(const float* __restrict__ sampled,
                                                   const float* __restrict__ parent,
                                                   unsigned* __restrict__ ghist) {
    __shared__ unsigned shist[NBINS];          // 32 KB
    __shared__ float    tiles[2][TILE_FLOATS]; // 8 KB
    const int tid = threadIdx.x;
    const int n = blockIdx.x, b = blockIdx.y;

    for (int i = tid; i < NBINS; i += 256) shist[i] = 0u;

    const float pp = parent[b * N_ + n];
    const float* gbase = sampled + ((size_t)b * N_ + (size_t)n) * (size_t)V_;
    const unsigned lds0   = (unsigned)(uintptr_t)(&tiles[0][0]); // low 32 bits = LDS offset
    const unsigned lane16 = (unsigned)tid * 16u;
    __syncthreads();

    // Prologue: tiles 0 and 1 in flight
    async_load_b128(lds0 + lane16, 0u * TILE_BYTES + lane16, gbase);
    async_load_b128(lds0 + TILE_BYTES + lane16, 1u * TILE_BYTES + lane16, gbase);

    for (int t = 0; t < NTILES; ++t) {
        if (t + 1 < NTILES) asm volatile("s_wait_asynccnt 1" ::: "memory");
        else                asm volatile("s_wait_asynccnt 0" ::: "memory");

        float4 v = *reinterpret_cast<float4*>(&tiles[t & 1][tid * 4]);
        unsigned b0 = __float_as_uint(v.x * pp) >> BIN_SHIFT;
        unsigned b1 = __float_as_uint(v.y * pp) >> BIN_SHIFT;
        unsigned b2 = __float_as_uint(v.z * pp) >> BIN_SHIFT;
        unsigned b3 = __float_as_uint(v.w * pp) >> BIN_SHIFT;
        atomicAdd(&shist[b0], 1u);
        atomicAdd(&shist[b1], 1u);
        atomicAdd(&shist[b2], 1u);
        atomicAdd(&shist[b3], 1u);

        // anti-dependency fence: our LDS reads must retire before this lane's
        // next async load overwrites the same buffer slot
        asm volatile("s_wait_dscnt 0" ::: "memory");
        if (t + 2 < NTILES)
            async_load_b128(lds0 + (unsigned)(t & 1) * TILE_BYTES + lane16,
                            (unsigned)(t + 2) * TILE_BYTES + lane16, gbase);
    }
    __syncthreads();

    unsigned* gh = ghist + (size_t)b * NBINS;
    for (int i = tid; i < NBINS; i += 256)
        if (shist[i]) atomicAdd(&gh[i], shist[i]);
}

// ---------------------------------------------------------------------------
// Pass 2 (tiny): per batch row, find the largest bin t with suffix-count >= K.
// ---------------------------------------------------------------------------
__global__ __launch_bounds__(256) void scan_kernel(const unsigned* __restrict__ ghist,
                                                   unsigned* __restrict__ thresh) {
    const int b = blockIdx.x, tid = threadIdx.x;
    const unsigned* h = ghist + (size_t)b * NBINS;
    __shared__ unsigned seg[256];
    __shared__ unsigned suf[256];
    const int SEG = NBINS / 256; // 32
    const int lo = tid * SEG;
    unsigned s = 0;
    for (int i = 0; i < SEG; ++i) s += h[lo + i];
    seg[tid] = s;
    __syncthreads();
    if (tid == 0) {
        unsigned acc = 0;
        for (int i = 255; i >= 0; --i) { suf[i] = acc; acc += seg[i]; }
    }
    __syncthreads();
    const unsigned after = suf[tid]; // count in segments strictly above mine
    if (after < K_ && after + seg[tid] >= K_) {
        unsigned c = after;
        for (int bin = lo + SEG - 1; bin >= lo; --bin) {
            c += h[bin];
            if (c >= K_) { thresh[b] = (unsigned)bin; break; }
        }
    }
}

// ---------------------------------------------------------------------------
// Pass 3: re-stream, collect all elements whose bin >= threshold.
// ---------------------------------------------------------------------------
__global__ __launch_bounds__(256) void collect_kernel(const float* __restrict__ sampled,
                                                      const float* __restrict__ parent,
                                                      const unsigned* __restrict__ thresh,
                                                      unsigned* __restrict__ counters,
                                                      uint2* __restrict__ cands) {
    const int tid = threadIdx.x;
    const int n = blockIdx.x, b = blockIdx.y;
    const float pp = parent[b * N_ + n];
    const unsigned th = thresh[b];
    const float* src = sampled + ((size_t)b * N_ + (size_t)n) * (size_t)V_;
    const unsigned rowbase = (unsigned)n * (unsigned)V_;

    for (int i = tid * 4; i < V_; i += 1024) {
        __builtin_prefetch(src + i + 16384, 0, 1); // -> global_prefetch_b8
        float4 v = *reinterpret_cast<const float4*>(src + i);
        float p0 = v.x * pp, p1 = v.y * pp, p2 = v.z * pp, p3 = v.w * pp;
#define EMIT(pv, off)                                                           \
        do {                                                                    \
            unsigned bits = __float_as_uint(pv);                                \
            if ((bits >> BIN_SHIFT) >= th) {                                    \
                unsigned pos = atomicAdd(&counters[b], 1u);                     \
                if (pos < CAP)                                                  \
                    cands[(size_t)b * CAP + pos] =                              \
                        make_uint2(bits, rowbase + (unsigned)i + (off));        \
            }                                                                   \
        } while (0)
        EMIT(p0, 0u); EMIT(p1, 1u); EMIT(p2, 2u); EMIT(p3, 3u);
#undef EMIT
    }
}

// ---------------------------------------------------------------------------
// Pass 4 (tiny): per batch row, 64 iterations of block-wide argmax over the
// candidate set; emit token id / prob / parent index in descending order.
// ---------------------------------------------------------------------------
__global__ __launch_bounds__(256) void final_kernel(const uint2* __restrict__ cands,
                                                    const unsigned* __restrict__ counters,
                                                    float* __restrict__ out) {
    __shared__ unsigned cvals[CAP];
    __shared__ unsigned cidx[CAP];
    __shared__ unsigned long long sred[256];
    const int tid = threadIdx.x;
    const int b = blockIdx.x;
    unsigned ncand = counters[b];
    if (ncand > CAP) ncand = CAP;
    for (int i = tid; i < CAP; i += 256) {
        if ((unsigned)i < ncand) {
            uint2 c = cands[(size_t)b * CAP + i];
            cvals[i] = c.x; cidx[i] = c.y;
        } else {
            cvals[i] = 0u; cidx[i] = 0u;
        }
    }
    __syncthreads();
    for (int k = 0; k < K_; ++k) {
        unsigned long long best = 0ull;
        for (int s = tid; s < CAP; s += 256) {
            unsigned long long key = ((unsigned long long)cvals[s] << 32) | (unsigned)s;
            if (key > best) best = key;
        }
        sred[tid] = best;
        __syncthreads();
        for (int step = 128; step > 0; step >>= 1) {
            if (tid < step && sred[tid + step] > sred[tid]) sred[tid] = sred[tid + step];
            __syncthreads();
        }
        if (tid == 0) {
            unsigned long long w = sred[0];
            unsigned slot = (unsigned)(w & 0xFFFFFFFFu);
            unsigned bits = (unsigned)(w >> 32);
            unsigned flat = cidx[slot];
            out[b * K_ + k]            = (float)(flat % (unsigned)V_); // token id
            out[BK_ + b * K_ + k]      = __uint_as_float(bits);        // prob
            out[2 * BK_ + b * K_ + k]  = (float)(flat / (unsigned)V_); // parent leaf
            cvals[slot] = 0u;
        }
        __syncthreads();
    }
}

extern "C" void kernel_launch(void* const* d_in, const int* in_sizes, int n_in,
                              void* d_out, int out_size, void* d_ws, size_t ws_size,
                              hipStream_t stream) {
    (void)in_sizes; (void)n_in; (void)out_size; (void)ws_size;
    const float* sampled = (const float*)d_in[0];   // [B, N, V] fp32
    const float* parent  = (const float*)d_in[1];   // [B, N] fp32
    float* out = (float*)d_out;                     // 3 * B * K floats

    uint8_t* ws = (uint8_t*)d_ws;
    unsigned* hist     = (unsigned*)(ws);
    unsigned* counters = (unsigned*)(ws + CNT_OFF);
    unsigned* thresh   = (unsigned*)(ws + THR_OFF);
    uint2*    cands    = (uint2*)   (ws + CAND_OFF);

    hipMemsetAsync(ws, 0, HIST_BYTES + 128, stream);   // hist + counters + thresh
    hist_kernel   <<<dim3(N_, B_), 256, 0, stream>>>(sampled, parent, hist);
    scan_kernel   <<<B_,           256, 0, stream>>>(hist, thresh);
    collect_kernel<<<dim3(N_, B_), 256, 0, stream>>>(sampled, parent, thresh, counters, cands);
    final_kernel  <<<B_,           256, 0, stream>>>(cands, counters, out);
}